// ModelNew_80908593922634
// MI455X (gfx1250) — compile-verified
//
#include <hip/hip_runtime.h>
#include <float.h>

typedef __attribute__((ext_vector_type(2))) float v2f;
typedef __attribute__((ext_vector_type(8))) float v8f;
typedef __attribute__((ext_vector_type(4))) unsigned int u32x4;
typedef __attribute__((ext_vector_type(4))) int i32x4;
typedef __attribute__((ext_vector_type(8))) int i32x8;

#if defined(__has_builtin)
#  if __has_builtin(__builtin_amdgcn_tensor_load_to_lds) && \
      __has_builtin(__builtin_amdgcn_s_wait_tensorcnt)
#    define HAVE_TDM 1
#  endif
#endif
#ifndef HAVE_TDM
#  define HAVE_TDM 0
#endif

// ---- problem constants -------------------------------------------------
#define IN_C   3
#define IN_D   24
#define IN_H   128
#define IN_W   128
#define OC     16
#define OUT_D  22
#define OUT_H  126
#define OUT_W  126
#define TILES_W 8                  // ceil(126/16)
#define WAVES_PER_BLOCK 4
#define WTOT   (OC * 81)           // 1296 weight elements

#define S_H   (IN_W)               // 128
#define S_D   (IN_H * IN_W)        // 16384
#define S_C   (IN_D * S_D)         // 393216
#define S_B   (IN_C * S_C)         // 1179648

// K reordered: kd outermost. Within a kd-group, j = c*9 + kh*3 + kw (27 vals,
// padded to 28 -> 7 chunks of K=4 per group, 21 WMMAs per depth).
#define JGRP  27
#define CHG   7                    // chunks per kd-group

// j -> element offset within one input plane-origin (no depth component)
__device__ __forceinline__ int joff(int j) {
    int c  = j / 9;
    int r  = j - c * 9;
    int kh = r / 3;
    int kw = r - kh * 3;
    return c * S_C + kh * S_H + kw;
}

// Load one kd-group's A fragments (7 chunks) from a given input plane base.
__device__ __forceinline__ void load_group(v2f (&g)[CHG],
                                           const float* __restrict__ base,
                                           const int (&o0)[CHG], const int (&o1)[CHG]) {
    #pragma unroll
    for (int i = 0; i < CHG; ++i) {
        g[i].x = base[o0[i]];
        g[i].y = base[o1[i]];
    }
}

// One output depth: acc = Gd0*B0 + Gd1*B1 + Gd2*B2 (21 f32 WMMAs), fold min.
__device__ __forceinline__ void gemm_step(v8f& vmin,
        const v2f (&G0)[CHG], const v2f (&G1)[CHG], const v2f (&G2)[CHG],
        const v2f (&B0)[CHG], const v2f (&B1)[CHG], const v2f (&B2)[CHG]) {
    v8f acc;
    #pragma unroll
    for (int r = 0; r < 8; ++r) acc[r] = 0.0f;
    #pragma unroll
    for (int i = 0; i < CHG; ++i)
        acc = __builtin_amdgcn_wmma_f32_16x16x4_f32(false, G0[i], false, B0[i],
                                                    (short)0, acc, false, false);
    #pragma unroll
    for (int i = 0; i < CHG; ++i)
        acc = __builtin_amdgcn_wmma_f32_16x16x4_f32(false, G1[i], false, B1[i],
                                                    (short)0, acc, false, false);
    #pragma unroll
    for (int i = 0; i < CHG; ++i)
        acc = __builtin_amdgcn_wmma_f32_16x16x4_f32(false, G2[i], false, B2[i],
                                                    (short)0, acc, false, false);
    #pragma unroll
    for (int r = 0; r < 8; ++r) vmin[r] = fminf(vmin[r], acc[r]);
}

__global__ __launch_bounds__(WAVES_PER_BLOCK * 32)
void conv3d_min_softmax_wmma(const float* __restrict__ x,
                             const float* __restrict__ wgt,
                             float* __restrict__ out) {
#if HAVE_TDM
    __shared__ float lds_w[WTOT];
#endif

    const int lane = threadIdx.x & 31;
    const int wave = threadIdx.x >> 5;
    const int tile = blockIdx.x * WAVES_PER_BLOCK + wave;    // uniform per wave

    // 8064 tiles = 4 * 2016 blocks exactly; no wave is ever out of range.
    // tile -> (batch, output row h, w-tile)
    const int b   = tile / (OUT_H * TILES_W);
    const int rem = tile - b * (OUT_H * TILES_W);
    const int h   = rem / TILES_W;
    const int tw  = rem - h * TILES_W;
    const int w0  = tw * 16;

    const int half = lane >> 4;    // K-pair select within chunk
    const int m    = lane & 15;    // A: M (pixel in tile); B/C/D: N (oc)
    // Clamp pixel for the partial tile (last tile has 14 valid pixels);
    // clamped lanes duplicate pixel 125 and are masked at the store.
    const int wpx  = (w0 + m < OUT_W) ? (w0 + m) : (OUT_W - 1);
    const int oc   = m;

#if HAVE_TDM
    // ---- Stage the full weight tensor into LDS via the Tensor Data Mover.
    // 1-D descriptor: 1296 fp32 elements, tile == tensor. Every wave issues
    // its own TDM op and waits on its own TENSORcnt; concurrent duplicate
    // writes of identical bytes are benign, so no block barrier is needed.
    {
        unsigned long long ga = (unsigned long long)(const void*)wgt;
        unsigned int ldsOff   = (unsigned int)(unsigned long long)(void*)&lds_w[0];

        u32x4 g0;
        g0[0] = 1u;                                   // count=1 (valid user D#)
        g0[1] = ldsOff;                               // lds_addr (bytes)
        g0[2] = (unsigned int)ga;                     // global_addr[31:0]
        g0[3] = (unsigned int)((ga >> 32) & 0x01FFFFFFull) | (2u << 30); // [56:32] | type=2

        i32x8 g1;
        g1[0] = (int)(2u << 16);                      // data_size=4B; mask=0
        g1[1] = (int)((unsigned)WTOT << 16);          // tensor_dim0[15:0] in [31:16]
        g1[2] = (int)(1u << 16);                      // tensor_dim0 hi=0; tensor_dim1=1
        g1[3] = (int)((unsigned)WTOT << 16);          // tensor_dim1 hi=0; tile_dim0=1296
        g1[4] = 1;                                    // tile_dim1=1; tile_dim2=0
        g1[5] = WTOT;                                 // tensor_dim0_stride (low 32)
        g1[6] = 0;
        g1[7] = 0;

        i32x4 gz = {0, 0, 0, 0};
#if __clang_major__ >= 23
        i32x8 gz8 = {0, 0, 0, 0, 0, 0, 0, 0};
        __builtin_amdgcn_tensor_load_to_lds(g0, g1, gz, gz, gz8, 0);
#else
        __builtin_amdgcn_tensor_load_to_lds(g0, g1, gz, gz, 0);
#endif
        __builtin_amdgcn_s_wait_tensorcnt(0);
    }
#endif

    // weight accessor: (kd-group g, slot j); zero for pad slot j >= 27
    auto wl = [&](int g, int j) -> float {
        if (j >= JGRP) return 0.0f;
        int c = j / 9;
        int r = j - c * 9;                            // kh*3 + kw
        int idx = oc * 81 + c * 27 + g * 9 + r;
#if HAVE_TDM
        return lds_w[idx];                            // ds_load_b32
#else
        return wgt[idx];
#endif
    };

    // ---- B fragments per kd-group (zero-padded j=27) --------------------
    // B layout (4x16): lane = N(oc); v.x holds K-slot 4i+2*half, v.y slot+1.
    v2f B0[CHG], B1[CHG], B2[CHG];
    #pragma unroll
    for (int i = 0; i < CHG; ++i) {
        const int j0 = 4 * i + 2 * half;
        const int j1 = j0 + 1;
        B0[i].x = wl(0, j0);  B0[i].y = wl(0, j1);
        B1[i].x = wl(1, j0);  B1[i].y = wl(1, j1);
        B2[i].x = wl(2, j0);  B2[i].y = wl(2, j1);
    }

    // ---- Per-lane plane-relative A offsets (group/depth invariant) -------
    int o0[CHG], o1[CHG];
    #pragma unroll
    for (int i = 0; i < CHG; ++i) {
        int j0 = 4 * i + 2 * half;
        int j1 = j0 + 1;
        if (j0 > JGRP - 1) j0 = JGRP - 1;   // pad slots: B is zero there
        if (j1 > JGRP - 1) j1 = JGRP - 1;
        o0[i] = joff(j0);
        o1[i] = joff(j1);
    }

    // Per-lane base pointer (no depth component).
    const float* xp = x + (long)b * S_B + (long)h * S_H + wpx;

    // ---- Rolling 3-plane register pipeline over output depth ------------
    v8f vmin;
    #pragma unroll
    for (int r = 0; r < 8; ++r) vmin[r] = FLT_MAX;

    v2f A0[CHG], A1[CHG], A2[CHG];
    load_group(A0, xp + 0 * S_D, o0, o1);   // plane d
    load_group(A1, xp + 1 * S_D, o0, o1);   // plane d+1
    load_group(A2, xp + 2 * S_D, o0, o1);   // plane d+2

    for (int d3 = 0; d3 < OUT_D - 1; d3 += 3) {     // d3 = 0,3,...,18
        gemm_step(vmin, A0, A1, A2, B0, B1, B2);    // depth d3
        load_group(A0, xp + (d3 + 3) * S_D, o0, o1);
        gemm_step(vmin, A1, A2, A0, B0, B1, B2);    // depth d3+1
        load_group(A1, xp + (d3 + 4) * S_D, o0, o1);
        gemm_step(vmin, A2, A0, A1, B0, B1, B2);    // depth d3+2
        load_group(A2, xp + (d3 + 5) * S_D, o0, o1);
    }
    gemm_step(vmin, A0, A1, A2, B0, B1, B2);        // depth 21: planes 21,22,23

    // ---- Softmax over channels (N = lanes within each 16-lane half) -----
    const long ocStride = (long)OUT_H * OUT_W;      // 15876
    float* outBase = out + ((long)b * OC + oc) * ocStride + (long)h * OUT_W;

    #pragma unroll
    for (int r = 0; r < 8; ++r) {
        float v = vmin[r];
        float mx = v;
        #pragma unroll
        for (int mask = 1; mask < 16; mask <<= 1)
            mx = fmaxf(mx, __shfl_xor(mx, mask, 32));
        float e = __expf(v - mx);
        float s = e;
        #pragma unroll
        for (int mask = 1; mask < 16; mask <<= 1)
            s += __shfl_xor(s, mask, 32);
        float res = e / s;

        const int pix = r + 8 * half;               // pixel index within tile
        const int wo  = w0 + pix;
        if (wo < OUT_W)
            outBase[wo] = res;
    }
}

extern "C" void kernel_launch(void* const* d_in, const int* in_sizes, int n_in,
                              void* d_out, int out_size, void* d_ws, size_t ws_size,
                              hipStream_t stream) {
    const float* x = (const float*)d_in[0];
    const float* w = (const float*)d_in[1];
    float* out = (float*)d_out;

    const int total_tiles = 8 * OUT_H * TILES_W;    // 8064 waves, 2016 blocks
    const int blocks = (total_tiles + WAVES_PER_BLOCK - 1) / WAVES_PER_BLOCK;
    conv3d_min_softmax_wmma<<<blocks, WAVES_PER_BLOCK * 32, 0, stream>>>(x, w, out);
}